// BlockDiagonal_52922587021428
// MI455X (gfx1250) — compile-verified
//
#include <hip/hip_runtime.h>
#include <math.h>

// Block-diagonal linear: out[r, b*64+i] = sum_j x[r, b*64+j] * tanh(W[b*64+i, b*64+j])
// 64 independent [8192x64] @ [64x64]^T GEMMs. HBM-bound (~270MB @ 23.3TB/s ~= 11.5us).
// f32 WMMA (V_WMMA_F32_16X16X4_F32) for the math; GLOBAL_LOAD_ASYNC_TO_LDS_B128 +
// double buffering for the x stream (the CDNA5-idiomatic global->LDS path).

typedef float v2f __attribute__((ext_vector_type(2)));
typedef float v8f __attribute__((ext_vector_type(8)));
typedef int   v4i __attribute__((vector_size(16)));
typedef __attribute__((address_space(1))) v4i* gptr_v4i;   // global
typedef __attribute__((address_space(3))) v4i* lptr_v4i;   // LDS

#define LSZ        4096   // layer size
#define ROWS       8192
#define BLK        64     // block size
#define NBLK       64     // number of diagonal blocks
#define TILE_ROWS  32     // rows staged per iteration (2 row-tiles x 4 col-tiles = 8 waves)
#define XPAD       68     // Xs row pad: A-frag column reads conflict-free, rows 16B-aligned
#define BPAD       72     // Bs row pad: lane-half rows j/j+2 land in disjoint bank groups
#define ITERS      8      // row tiles per workgroup -> 256 rows per WG

#if defined(__AMDGCN__) && __has_builtin(__builtin_amdgcn_global_load_async_to_lds_b128)
#define USE_ASYNC 1
#else
#define USE_ASYNC 0
#endif

__device__ __forceinline__ void wait_async_all()
{
#if USE_ASYNC
#if __has_builtin(__builtin_amdgcn_s_wait_asynccnt)
    __builtin_amdgcn_s_wait_asynccnt(0);
#else
    asm volatile("s_wait_asynccnt 0x0" ::: "memory");
#endif
#endif
}

// Stage one 32x64 f32 tile: 512 x 16B transfers, 2 per thread, coalesced.
// xg points at x[r0*LSZ + b*64]. Async mode: direct global->LDS, no VGPR round-trip.
__device__ __forceinline__ void stage_tile(float (*__restrict__ Xbuf)[XPAD],
                                           const float* __restrict__ xg, int tid)
{
    const int rr0 = tid >> 4;          // 0..15
    const int c4  = (tid & 15) * 4;    // 0,4,...,60
#pragma unroll
    for (int h = 0; h < 2; ++h) {
        const int rr = rr0 + h * 16;
        const float* gp = xg + (size_t)rr * LSZ + c4;
        float* lp = &Xbuf[rr][c4];
#if USE_ASYNC
        __builtin_amdgcn_global_load_async_to_lds_b128(
            (gptr_v4i)gp, (lptr_v4i)lp, /*offset=*/0, /*cpol=*/0);
#else
        *(float4*)lp = *(const float4*)gp;
#endif
    }
}

__global__ __launch_bounds__(256)
void bdiag_wmma_kernel(const float* __restrict__ x,
                       const float* __restrict__ W,
                       float* __restrict__ out)
{
    // Bs[j][i] = tanh(W[b*64+i, b*64+j])  (pre-transposed: j = K dim, i = output col)
    __shared__ float Bs[BLK][BPAD];
    __shared__ float Xs[2][TILE_ROWS][XPAD];   // double-buffered x tiles

    const int b   = blockIdx.y;
    const int b64 = b * BLK;
    const int tid = threadIdx.x;

    const int rowBase   = blockIdx.x * (TILE_ROWS * ITERS);
    const float* xblk   = x + (size_t)rowBase * LSZ + b64;

    // Prefetch tile 0 while the tanh preamble runs.
    stage_tile(Xs[0], xblk, tid);

    // Load W tile (coalesced), tanh, store transposed into LDS. Once per WG.
    for (int e = tid; e < BLK * BLK; e += 256) {
        const int i = e >> 6;   // row within block = output column index
        const int j = e & 63;   // col within block = K index
        Bs[j][i] = tanhf(W[(size_t)(b64 + i) * LSZ + (b64 + j)]);
    }

    const int lane    = tid & 31;
    const int wave    = tid >> 5;          // 0..7
    const int waveRow = (wave >> 2) * 16;  // 0 or 16: row-tile within the 32-row chunk
    const int cbase   = (wave & 3) * 16;   // 0/16/32/48: col-tile within the 64-col block
    const int lm      = lane & 15;
    const int hi      = lane >> 4;         // 0: lanes 0-15, 1: lanes 16-31
    const int koff    = hi * 2;            // A/B fragments: upper lanes carry K+2,K+3
    const int arow    = waveRow + lm;      // A: both lane halves carry the same M rows

    for (int it = 0; it < ITERS; ++it) {
        // Publish current buffer (and Bs on the first iteration). The barrier also
        // guarantees all waves finished reading the other buffer last iteration.
        wait_async_all();
        __syncthreads();

        // Prefetch next tile into the other buffer, overlapping with WMMA below.
        if (it + 1 < ITERS)
            stage_tile(Xs[(it + 1) & 1], xblk + (size_t)(it + 1) * TILE_ROWS * LSZ, tid);

        const float (*__restrict__ Xc)[XPAD] = Xs[it & 1];
        const int r0 = rowBase + it * TILE_ROWS;

        // 16x16 output tile per wave: K=64 via 16 chained f32 WMMAs (K-step 4).
        v8f acc = {};
#pragma unroll
        for (int k = 0; k < BLK; k += 4) {
            v2f a, bf;
            // A 16x4 f32 layout: VGPR0 = K=k+koff, VGPR1 = K=k+koff+1 (lane half picks koff)
            a.x = Xc[arow][k + koff];
            a.y = Xc[arow][k + koff + 1];
            // B 4x16 f32 layout mirrors A: lane = N, lane half picks K+0/1 vs K+2/3
            bf.x = Bs[k + koff][cbase + lm];
            bf.y = Bs[k + koff + 1][cbase + lm];
            acc = __builtin_amdgcn_wmma_f32_16x16x4_f32(
                /*neg_a=*/false, a, /*neg_b=*/false, bf,
                /*c_mod=*/(short)0, acc, /*reuse_a=*/false, /*reuse_b=*/false);
        }

        // Store D (16x16 f32 C/D layout): VGPR v -> row v + 8*hi, col = lane&15.
        const size_t outBase = (size_t)(r0 + waveRow + hi * 8) * LSZ + (b64 + cbase + lm);
#pragma unroll
        for (int v = 0; v < 8; ++v) {
            out[outBase + (size_t)v * LSZ] = acc[v];
        }
    }
}

extern "C" void kernel_launch(void* const* d_in, const int* in_sizes, int n_in,
                              void* d_out, int out_size, void* d_ws, size_t ws_size,
                              hipStream_t stream)
{
    const float* x = (const float*)d_in[0];   // [8192, 4096] f32
    const float* W = (const float*)d_in[1];   // [4096, 4096] f32
    // d_in[2] is the boolean block-diagonal mask; its structure is static, so it is
    // baked into the addressing (only diagonal 64x64 tiles of W are read).
    float* out = (float*)d_out;               // [8192, 4096] f32

    dim3 grid(ROWS / (TILE_ROWS * ITERS), NBLK);  // (32, 64)
    bdiag_wmma_kernel<<<grid, dim3(256), 0, stream>>>(x, W, out);
}